// PhotoreceptorLayer_17274358464636
// MI455X (gfx1250) — compile-verified
//
#include <hip/hip_runtime.h>

// ---------------------------------------------------------------------------
// PhotoreceptorLayer forward: elementwise over 2048x2048 fp32 mosaic.
// Bandwidth-bound: 168 MB traffic/launch vs ~0.5 GFLOP -> optimize data path.
// MI455X plan: b128 loads, wave32 blocks, async-DMA (GLOBAL_LOAD_ASYNC_TO_LDS)
// double-buffered staging with s_wait_asynccnt; RT cache hints so the whole
// 168 MB working set stays resident in the 192 MB L2 across graph replays.
// ---------------------------------------------------------------------------

#define TPB 256         // 8 wave32 waves per block
#define VEC 4           // 4 f32 per lane -> b128 transactions
#define TILE (TPB * VEC)   // 1024 elements per tile
#define NLDS 8          // 8 float streams staged through LDS (rod_mask direct)

#if defined(__gfx1250__) && __has_builtin(__builtin_amdgcn_global_load_async_to_lds_b128)
#define HAVE_ASYNC 1
#else
#define HAVE_ASYNC 0
#endif

__device__ __forceinline__ float frcp(float x) { return __builtin_amdgcn_rcpf(x); }

__device__ __forceinline__ float ftanh(float x) {
#if defined(__gfx1250__) && __has_builtin(__builtin_amdgcn_tanhf)
  return __builtin_amdgcn_tanhf(x);       // v_tanh_f32 (CDNA5 TRANS op)
#else
  float e = __expf(2.0f * x);             // tanh(x) = (e^2x - 1)/(e^2x + 1)
  return (e - 1.0f) * frcp(e + 1.0f);
#endif
}

// Per-element photoreceptor model (matches reference math in fp32).
__device__ __forceinline__ float photoreceptor(float img, float nz, float ad,
                                               float fs, float g, float tau,
                                               float pk, float bw, int rm) {
  float light = fmaxf(fmaf(nz, 0.01f, img), 0.0f);            // image + noise*0.01
  float d = 530.0f - pk;                                      // WAVELENGTH - peak
  float spectral = __expf(-(d * d) * (0.5f * frcp(bw * bw)));
  light *= spectral;
  float target = frcp(1.0f + light);                          // 1/(1+light)
  float adapt = fmaf(0.1f, target - ad, ad);                  // ADAPT_RATE=0.1, DT=1
  adapt = fminf(fmaxf(adapt, 0.1f), 10.0f);
  float alpha = __expf(-frcp(tau));                           // exp(-DT/tau)
  float filt = fmaf(alpha, fs - light, light);                // a*fs + (1-a)*light
  float resp = ftanh(g * __logf(fmaf(filt, adapt, 1.0f)));    // tanh(g*log1p(f*a))
  float llg = 10.0f * frcp(1.0f + img + 1e-8f);               // scotopic gain
  return (rm > 0) ? resp * llg : resp;
}

#if HAVE_ASYNC
// Builtin signature (from hipcc diagnostic): (v4i addrspace(1)*, v4i addrspace(3)*, Ii, Ii)
typedef int v4i_t __attribute__((ext_vector_type(4)));
typedef __attribute__((address_space(1))) v4i_t* g_v4i_ptr;
typedef __attribute__((address_space(3))) v4i_t* l_v4i_ptr;

__device__ __forceinline__ void async_copy16(const void* gsrc, void* ldst) {
  // 16B/lane DMA: global -> LDS, tracked by ASYNCcnt, no VGPR staging.
  __builtin_amdgcn_global_load_async_to_lds_b128(
      (g_v4i_ptr)gsrc, (l_v4i_ptr)ldst, /*offset=*/0, /*cpol=*/0);
}

template <int N>
__device__ __forceinline__ void wait_async() {
#if __has_builtin(__builtin_amdgcn_s_wait_asynccnt)
  __builtin_amdgcn_s_wait_asynccnt(N);
  asm volatile("" ::: "memory");          // keep LDS reads below the wait
#else
  asm volatile("s_wait_asynccnt %0" ::"i"(N) : "memory");
#endif
}
#endif  // HAVE_ASYNC

__global__ void __launch_bounds__(TPB) photoreceptor_kernel(
    const float* __restrict__ img, const float* __restrict__ nz,
    const float* __restrict__ ad, const float* __restrict__ fs,
    const float* __restrict__ gn, const float* __restrict__ tu,
    const float* __restrict__ pk, const float* __restrict__ bw,
    const int* __restrict__ rm, float* __restrict__ out, int ntiles) {
  const int t = threadIdx.x;

#if HAVE_ASYNC
  // 2 buffers x 8 streams x 4KB = 64KB LDS (WGP pool is 320KB -> 4+ blocks/WGP).
  __shared__ __align__(16) unsigned char lds[2][NLDS][TILE * 4];
  const float* const gsrc[NLDS] = {img, nz, ad, fs, gn, tu, pk, bw};

  auto issue = [&](int b, int tile) {
    const size_t base = (size_t)tile * TILE + (size_t)t * VEC;
#pragma unroll
    for (int k = 0; k < NLDS; ++k)
      async_copy16(gsrc[k] + base, &lds[b][k][(size_t)t * 16]);
  };

  int buf = 0;
  int tile = blockIdx.x;
  if (tile < ntiles) issue(buf, tile);          // prologue prefetch

  for (; tile < ntiles; tile += gridDim.x) {
    const int nxt = tile + (int)gridDim.x;
    if (nxt < ntiles) {
      issue(buf ^ 1, nxt);                      // prefetch next tile (8 DMAs)
      wait_async<NLDS>();                       // current tile's 8 DMAs done
    } else {
      wait_async<0>();
    }

    float4 v[NLDS];
#pragma unroll
    for (int k = 0; k < NLDS; ++k)
      v[k] = *reinterpret_cast<const float4*>(&lds[buf][k][(size_t)t * 16]);
    const int4 vr = ((const int4*)rm)[(size_t)tile * TPB + t];

    float4 r;
    float* rp = (float*)&r;
    const int* rmp = (const int*)&vr;
#pragma unroll
    for (int i = 0; i < VEC; ++i) {
      rp[i] = photoreceptor(((const float*)&v[0])[i], ((const float*)&v[1])[i],
                            ((const float*)&v[2])[i], ((const float*)&v[3])[i],
                            ((const float*)&v[4])[i], ((const float*)&v[5])[i],
                            ((const float*)&v[6])[i], ((const float*)&v[7])[i],
                            rmp[i]);
    }
    ((float4*)out)[(size_t)tile * TPB + t] = r;  // global_store_b128
    buf ^= 1;
  }
#else
  // Fallback: pure b128 streaming loads (still HBM-bandwidth optimal).
  for (int tile = blockIdx.x; tile < ntiles; tile += gridDim.x) {
    const size_t b4 = (size_t)tile * TPB + t;   // index in float4 units
    float4 vimg = ((const float4*)img)[b4];
    float4 vnz  = ((const float4*)nz)[b4];
    float4 vad  = ((const float4*)ad)[b4];
    float4 vfs  = ((const float4*)fs)[b4];
    float4 vgn  = ((const float4*)gn)[b4];
    float4 vtu  = ((const float4*)tu)[b4];
    float4 vpk  = ((const float4*)pk)[b4];
    float4 vbw  = ((const float4*)bw)[b4];
    int4   vrm  = ((const int4*)rm)[b4];
    float4 r;
    float* rp = (float*)&r;
#pragma unroll
    for (int i = 0; i < VEC; ++i) {
      rp[i] = photoreceptor(((const float*)&vimg)[i], ((const float*)&vnz)[i],
                            ((const float*)&vad)[i], ((const float*)&vfs)[i],
                            ((const float*)&vgn)[i], ((const float*)&vtu)[i],
                            ((const float*)&vpk)[i], ((const float*)&vbw)[i],
                            ((const int*)&vrm)[i]);
    }
    ((float4*)out)[b4] = r;
  }
#endif
}

// Scalar tail for N % TILE (zero for 2048x2048, kept for robustness).
__global__ void __launch_bounds__(TPB) photoreceptor_tail(
    const float* __restrict__ img, const float* __restrict__ nz,
    const float* __restrict__ ad, const float* __restrict__ fs,
    const float* __restrict__ gn, const float* __restrict__ tu,
    const float* __restrict__ pk, const float* __restrict__ bw,
    const int* __restrict__ rm, float* __restrict__ out, int start, int total) {
  const int i = start + blockIdx.x * TPB + threadIdx.x;
  if (i < total)
    out[i] = photoreceptor(img[i], nz[i], ad[i], fs[i], gn[i], tu[i], pk[i],
                           bw[i], rm[i]);
}

extern "C" void kernel_launch(void* const* d_in, const int* in_sizes, int n_in,
                              void* d_out, int out_size, void* d_ws,
                              size_t ws_size, hipStream_t stream) {
  const float* img = (const float*)d_in[0];
  const float* nz  = (const float*)d_in[1];
  const float* ad  = (const float*)d_in[2];
  const float* fs  = (const float*)d_in[3];
  const float* gn  = (const float*)d_in[4];
  const float* tu  = (const float*)d_in[5];
  const float* pk  = (const float*)d_in[6];
  const float* bw  = (const float*)d_in[7];
  const int*   rm  = (const int*)d_in[8];
  float* out = (float*)d_out;

  const int n = in_sizes[0];                 // 2048*2048 = 4,194,304
  const int ntiles = n / TILE;               // 4096 full tiles
  const int rem = n - ntiles * TILE;

  if (ntiles > 0) {
    // 1024 blocks -> 4 tiles each: enough blocks to saturate all WGPs while
    // giving each block >=2 tiles so the double-buffered DMA pipeline overlaps.
    const int blocks = ntiles < 1024 ? ntiles : 1024;
    hipLaunchKernelGGL(photoreceptor_kernel, dim3(blocks), dim3(TPB), 0, stream,
                       img, nz, ad, fs, gn, tu, pk, bw, rm, out, ntiles);
  }
  if (rem > 0) {
    const int tb = (rem + TPB - 1) / TPB;
    hipLaunchKernelGGL(photoreceptor_tail, dim3(tb), dim3(TPB), 0, stream,
                       img, nz, ad, fs, gn, tu, pk, bw, rm, out, ntiles * TILE,
                       n);
  }
}